// EdgeConvLayer_49675591746183
// MI455X (gfx1250) — compile-verified
//
#include <hip/hip_runtime.h>
#include <hip/hip_bf16.h>
#include <stdint.h>

typedef float v2f __attribute__((ext_vector_type(2)));
typedef float v8f __attribute__((ext_vector_type(8)));

#define IN_CH   64
#define OUT_CH  64
#define KDIM    128      // 2*IN_CH
#define W1S     130      // LDS row stride for W1^T (even -> 8B aligned float2)
#define W2S     66       // LDS row stride for W2^T
#define H1S     66       // LDS row stride for h1 scratch
#define WAVES   3
#define TILE    16
#define KEY_NEG_INF 0x007FFFFFu

__device__ __forceinline__ unsigned fkey(float f) {
    unsigned u = __float_as_uint(f);
    return (u & 0x80000000u) ? ~u : (u | 0x80000000u);
}

__global__ void init_out_kernel(unsigned* __restrict__ out, int n) {
    int i = blockIdx.x * blockDim.x + threadIdx.x;
    if (i < n) out[i] = KEY_NEG_INF;
}

__global__ void finalize_out_kernel(float* __restrict__ outf, int n) {
    int i = blockIdx.x * blockDim.x + threadIdx.x;
    if (i < n) {
        unsigned u = __float_as_uint(outf[i]);
        float v;
        if (u == KEY_NEG_INF) v = 0.0f;                       // no incoming edges -> 0
        else v = (u & 0x80000000u) ? __uint_as_float(u ^ 0x80000000u)
                                   : __uint_as_float(~u);
        outf[i] = v;
    }
}

__global__ __launch_bounds__(32 * WAVES)
void edgeconv_wmma_kernel(const float* __restrict__ x,
                          const int*   __restrict__ ei,   // [2][E] int32
                          const float* __restrict__ W1,   // [128][64] row-major
                          const float* __restrict__ b1,   // [64]
                          const float* __restrict__ W2,   // [64][64] row-major
                          const float* __restrict__ b2,   // [64]
                          unsigned*    __restrict__ out,  // [N][64] as keys
                          int n_edges)
{
    __shared__ __align__(16) float sW1T[64 * W1S];          // W1^T : [n][k]
    __shared__ __align__(16) float sW2T[64 * W2S];          // W2^T : [n][k]
    __shared__ __align__(16) float sH1[WAVES][16 * H1S];    // per-wave h1 [m][n]

    const int tid = threadIdx.x;

    // Stage transposed weights into LDS (coalesced global reads)
    for (int idx = tid; idx < KDIM * OUT_CH; idx += blockDim.x) {
        int k = idx >> 6, n = idx & 63;
        sW1T[n * W1S + k] = W1[idx];
    }
    for (int idx = tid; idx < OUT_CH * OUT_CH; idx += blockDim.x) {
        int k = idx >> 6, n = idx & 63;
        sW2T[n * W2S + k] = W2[idx];
    }
    __syncthreads();

    const int lane = tid & 31;
    const int wave = tid >> 5;
    const int g  = lane >> 4;     // K-group (0/1) for A/B fragments
    const int nl = lane & 15;     // M index for A, N index for B/C
    const int*  srcp = ei;
    const int*  dstp = ei + n_edges;
    float* h1 = &sH1[wave][0];

    // per-lane bias for the 4 N-tiles (column n = nt*16 + nl)
    float bias1[4], bias2[4];
    #pragma unroll
    for (int nt = 0; nt < 4; ++nt) {
        bias1[nt] = b1[nt * 16 + nl];
        bias2[nt] = b2[nt * 16 + nl];
    }

    const int n_tiles = (n_edges + TILE - 1) / TILE;
    const int gwave   = blockIdx.x * WAVES + wave;
    const int nwaves  = gridDim.x * WAVES;

    for (int tile = gwave; tile < n_tiles; tile += nwaves) {
        const int ebase = tile * TILE;

        // ---- Gather A-matrix fragments into registers ----
        // Lane owns edge m=nl; K elements {4t+2g, 4t+2g+1}: t<16 -> x_i, t>=16 -> x_j-x_i
        int e = ebase + nl;
        if (e >= n_edges) e = n_edges - 1;       // clamp (stores are guarded)
        const int si = srcp[e];                  // j
        const int di = dstp[e];                  // i
        const float* xi = x + (size_t)di * IN_CH;
        const float* xj = x + (size_t)si * IN_CH;
        v2f ai[16], ad[16];
        #pragma unroll
        for (int t = 0; t < 16; ++t) {
            int c = 4 * t + 2 * g;
            v2f vi = *(const v2f*)(xi + c);
            v2f vj = *(const v2f*)(xj + c);
            ai[t] = vi;
            ad[t] = vj - vi;
        }

        // ---- GEMM1: [16x128] x [128x64] -> 16x64, K in steps of 4 ----
        v8f acc[4] = {};
        #pragma unroll
        for (int k = 0; k < 16; ++k) {           // K = 0..63 (x_i part)
            int kb = 4 * k + 2 * g;
            #pragma unroll
            for (int nt = 0; nt < 4; ++nt) {
                v2f b = *(const v2f*)(&sW1T[(nt * 16 + nl) * W1S + kb]);
                acc[nt] = __builtin_amdgcn_wmma_f32_16x16x4_f32(
                    false, ai[k], false, b, (short)0, acc[nt], false, false);
            }
        }
        #pragma unroll
        for (int k = 0; k < 16; ++k) {           // K = 64..127 (x_j - x_i part)
            int kb = 4 * k + 2 * g + 64;
            #pragma unroll
            for (int nt = 0; nt < 4; ++nt) {
                v2f b = *(const v2f*)(&sW1T[(nt * 16 + nl) * W1S + kb]);
                acc[nt] = __builtin_amdgcn_wmma_f32_16x16x4_f32(
                    false, ad[k], false, b, (short)0, acc[nt], false, false);
            }
        }

        // ---- bias + ReLU, redistribute via per-wave LDS ----
        #pragma unroll
        for (int nt = 0; nt < 4; ++nt) {
            #pragma unroll
            for (int r = 0; r < 8; ++r) {
                float h = fmaxf(acc[nt][r] + bias1[nt], 0.0f);
                int m = r + 8 * g;               // C layout: M = r + 8*(lane/16)
                h1[m * H1S + nt * 16 + nl] = h;
            }
        }
        asm volatile("s_wait_dscnt 0" ::: "memory");  // same-wave LDS RAW fence

        // ---- GEMM2: [16x64] x [64x64] -> 16x64 ----
        v8f acc2[4] = {};
        #pragma unroll
        for (int k = 0; k < 16; ++k) {
            int kb = 4 * k + 2 * g;
            v2f a = *(const v2f*)(&h1[nl * H1S + kb]);
            #pragma unroll
            for (int nt = 0; nt < 4; ++nt) {
                v2f b = *(const v2f*)(&sW2T[(nt * 16 + nl) * W2S + kb]);
                acc2[nt] = __builtin_amdgcn_wmma_f32_16x16x4_f32(
                    false, a, false, b, (short)0, acc2[nt], false, false);
            }
        }

        // ---- bias + scatter-max (monotonic u32 keys, native atomic max) ----
        #pragma unroll
        for (int r = 0; r < 8; ++r) {
            int m  = r + 8 * g;
            int e2 = ebase + m;
            if (e2 < n_edges) {
                int d = dstp[e2];
                unsigned* orow = out + (size_t)d * OUT_CH;
                #pragma unroll
                for (int nt = 0; nt < 4; ++nt) {
                    float v = acc2[nt][r] + bias2[nt];
                    atomicMax(&orow[nt * 16 + nl], fkey(v));
                }
            }
        }
    }
}

extern "C" void kernel_launch(void* const* d_in, const int* in_sizes, int n_in,
                              void* d_out, int out_size, void* d_ws, size_t ws_size,
                              hipStream_t stream) {
    const float* x  = (const float*)d_in[0];
    const int*   ei = (const int*)  d_in[1];
    const float* W1 = (const float*)d_in[2];
    const float* b1 = (const float*)d_in[3];
    const float* W2 = (const float*)d_in[4];
    const float* b2 = (const float*)d_in[5];

    const int n_edges = in_sizes[1] / 2;
    unsigned* outu = (unsigned*)d_out;

    init_out_kernel<<<(out_size + 255) / 256, 256, 0, stream>>>(outu, out_size);

    const int blocks = 1024;   // grid-stride over 100K tiles, 3 waves/block
    edgeconv_wmma_kernel<<<blocks, 32 * WAVES, 0, stream>>>(
        x, ei, W1, b1, W2, b2, outu, n_edges);

    finalize_out_kernel<<<(out_size + 255) / 256, 256, 0, stream>>>(
        (float*)d_out, out_size);
}